// Lip_Direction_35158602285766
// MI455X (gfx1250) — compile-verified
//
#include <hip/hip_runtime.h>
#include <hip/hip_bf16.h>
#include <math.h>

#define KDIM     26      // 20 lip + 6 pose
#define KPAD     28      // padded to multiple of 4 for 16x16x4 WMMA steps
#define NDIM     512
#define EPSV     1e-8f
#define NPAIR    351     // 26*27/2 upper-triangular Gram entries

typedef __attribute__((ext_vector_type(2))) float v2f;
typedef __attribute__((ext_vector_type(8))) float v8f;

// ---------------------------------------------------------------------------
// Kernel 1: CholeskyQR of the 512x26 matrix  w = [weight | pose_weight] + eps
// One workgroup of 512 threads; thread t owns row t.
//   G = w^T w  (351 packed upper entries, via wave reductions + LDS atomics)
//   G = R^T R  (Cholesky on thread 0)
//   q_row = w_row * R^-1  (forward substitution per thread)
// Writes Q^T into Qt[k*512 + n], rows 26..27 zero-padded.
// ---------------------------------------------------------------------------
__global__ __launch_bounds__(512) void qr_cholesky_kernel(
    const float* __restrict__ pose_w,   // (512, 6)
    const float* __restrict__ lip_w,    // (512, 20)
    float* __restrict__ Qt)             // (KPAD, 512)
{
    const int t    = threadIdx.x;
    const int lane = t & 31;

    float v[KDIM];
#pragma unroll
    for (int i = 0; i < 20; ++i) v[i]      = lip_w[t * 20 + i] + EPSV;
#pragma unroll
    for (int i = 0; i < 6;  ++i) v[20 + i] = pose_w[t * 6 + i] + EPSV;

    __shared__ float G[NPAIR];
    __shared__ float R[KDIM][KDIM];

    for (int i = t; i < NPAIR; i += 512) G[i] = 0.0f;
    __syncthreads();

    // Gram matrix: packed upper triangle, sequential idx matches (i, j>=i)
    int idx = 0;
#pragma unroll 1
    for (int i = 0; i < KDIM; ++i) {
#pragma unroll 1
        for (int j = i; j < KDIM; ++j) {
            float p = v[i] * v[j];
#pragma unroll
            for (int off = 16; off > 0; off >>= 1)
                p += __shfl_xor(p, off, 32);
            if (lane == 0) atomicAdd(&G[idx], p);
            ++idx;
        }
    }
    __syncthreads();

    // Cholesky G = R^T R (R upper triangular), single thread — trivial work
    if (t == 0) {
        int base = 0; // running offset of packed row i
        for (int i = 0; i < KDIM; ++i) {
            for (int j = i; j < KDIM; ++j) {
                float s = G[base + (j - i)];
                for (int k = 0; k < i; ++k) s -= R[k][i] * R[k][j];
                if (i == j) R[i][j] = sqrtf(fmaxf(s, 1e-30f));
                else        R[i][j] = s / R[i][i];
            }
            base += (KDIM - i);
        }
    }
    __syncthreads();

    // Forward substitution: q * R = v  ->  Q = w R^-1
    float q[KDIM];
#pragma unroll 1
    for (int j = 0; j < KDIM; ++j) {
        float s = v[j];
        for (int i = 0; i < j; ++i) s -= q[i] * R[i][j];
        q[j] = s / R[j][j];
    }

#pragma unroll
    for (int k = 0; k < KDIM; ++k) Qt[k * NDIM + t] = q[k];
    Qt[26 * NDIM + t] = 0.0f;   // K padding rows for the WMMA loop
    Qt[27 * NDIM + t] = 0.0f;
}

// ---------------------------------------------------------------------------
// Kernel 2: out(B,512) = input(B,26) @ Qt(26,512) via V_WMMA_F32_16X16X4_F32.
// One wave per 16x16 output tile; 7 K-steps of 4 (K padded 26->28).
// A frag (16x4 f32): lane l<16 holds A[l][k0],A[l][k0+1]; lane 16+l holds
//   A[l][k0+2],A[l][k0+3]  (2 VGPRs/lane).
// B frag (4x16 f32): VGPR0 = rows k0 (lanes 0-15) / k0+2 (lanes 16-31),
//   VGPR1 = rows k0+1 / k0+3, N = lane%16.
// C/D (16x16 f32): VGPR i -> M=i (lanes 0-15), M=8+i (lanes 16-31).
// ---------------------------------------------------------------------------
__global__ __launch_bounds__(256) void qgemm_wmma_kernel(
    const float* __restrict__ input,   // (B, 26)
    const float* __restrict__ Qt,      // (KPAD, 512)
    float* __restrict__ out,           // (B, 512)
    int n_row_tiles)                   // B / 16
{
    const int wid   = (blockIdx.x * blockDim.x + threadIdx.x) >> 5;
    const int lane  = threadIdx.x & 31;
    const int lhalf = lane >> 4;       // 0: lanes 0-15, 1: lanes 16-31
    const int l     = lane & 15;

    const int col_tile = wid & 31;             // 512/16 = 32 col tiles
    const int row_tile = wid >> 5;
    if (row_tile >= n_row_tiles) return;

    const int row0 = row_tile << 4;
    const int col0 = col_tile << 4;

    const float* arow = input + (size_t)(row0 + l) * KDIM;
    const float* bcol = Qt + col0 + l;

    v8f acc = {};

#pragma unroll
    for (int kk = 0; kk < KPAD / 4; ++kk) {
        const int k0 = kk * 4 + lhalf * 2;     // lane-dependent K base
        // A: clamp address, zero via multiply (keeps EXEC all-1s, no branches)
        const int kc0 = k0     < KDIM ? k0     : KDIM - 1;
        const int kc1 = k0 + 1 < KDIM ? k0 + 1 : KDIM - 1;
        const float m0 = (k0     < KDIM) ? 1.0f : 0.0f;
        const float m1 = (k0 + 1 < KDIM) ? 1.0f : 0.0f;
        v2f a;
        a.x = arow[kc0] * m0;
        a.y = arow[kc1] * m1;
        // B: Qt is padded to KPAD rows, always in range
        v2f b;
        b.x = bcol[(size_t)k0 * NDIM];
        b.y = bcol[(size_t)(k0 + 1) * NDIM];

        acc = __builtin_amdgcn_wmma_f32_16x16x4_f32(
            /*neg_a=*/false, a, /*neg_b=*/false, b,
            /*c_mod=*/(short)0, acc, /*reuse_a=*/false, /*reuse_b=*/false);
    }

    float* op = out + (size_t)row0 * NDIM + col0 + l;
#pragma unroll
    for (int i = 0; i < 8; ++i) {
        const int m = i + (lhalf << 3);        // VGPR i: M=i / M=8+i
        op[(size_t)m * NDIM] = acc[i];
    }
}

extern "C" void kernel_launch(void* const* d_in, const int* in_sizes, int n_in,
                              void* d_out, int out_size, void* d_ws, size_t ws_size,
                              hipStream_t stream) {
    const float* input  = (const float*)d_in[0];   // (B, 26)
    const float* pose_w = (const float*)d_in[1];   // (512, 6)
    const float* lip_w  = (const float*)d_in[2];   // (512, 20)
    float* out = (float*)d_out;
    float* Qt  = (float*)d_ws;                     // (28, 512) = 57344 B

    const int B = in_sizes[0] / KDIM;              // 131072
    const int n_row_tiles = B / 16;                // 8192
    const int total_waves = n_row_tiles * (NDIM / 16);   // 262144
    const int waves_per_block = 256 / 32;
    const int blocks = (total_waves + waves_per_block - 1) / waves_per_block;

    qr_cholesky_kernel<<<1, 512, 0, stream>>>(pose_w, lip_w, Qt);
    qgemm_wmma_kernel<<<blocks, 256, 0, stream>>>(input, Qt, out, n_row_tiles);
}